// DrugTargetPredictor_352187319175
// MI455X (gfx1250) — compile-verified
//
#include <hip/hip_runtime.h>
#include <cmath>

typedef __attribute__((ext_vector_type(2))) float v2f;
typedef __attribute__((ext_vector_type(8))) float v8f;

#define HIDDEN 32
#define DFEAT  10

// K1: h = x @ d_W1 + b1 ; out = h ; zero the 32-float mean accumulator.
__global__ void k1_node_linear(const float* __restrict__ x,
                               const float* __restrict__ W1,
                               const float* __restrict__ b1,
                               float* __restrict__ h,
                               float* __restrict__ outbuf,
                               float* __restrict__ acc,
                               long long total) {
  long long tid = (long long)blockIdx.x * blockDim.x + threadIdx.x;
  if (blockIdx.x == 0 && threadIdx.x < HIDDEN) acc[threadIdx.x] = 0.0f;
  if (tid >= total) return;
  long long i = tid >> 5;      // node
  int       j = (int)(tid & 31); // hidden dim
  const float* xi = x + i * DFEAT;
  float s = b1[j];
#pragma unroll
  for (int k = 0; k < DFEAT; ++k) s += xi[k] * W1[k * HIDDEN + j];
  h[tid] = s;
  outbuf[tid] = s;
}

// K2: out[dst[e]] += h[src[e]]  -- 8 lanes per edge, float4 per lane.
__global__ void k2_edge_scatter(const int* __restrict__ src,
                                const int* __restrict__ dst,
                                const float* __restrict__ h,
                                float* __restrict__ outbuf,
                                long long n_edges) {
  long long gid = (long long)blockIdx.x * blockDim.x + threadIdx.x;
  long long e = gid >> 3;
  if (e >= n_edges) return;
  int q = (int)(gid & 7) * 4;                 // column base 0,4,...,28
  long long s = (long long)src[e];
  long long d = (long long)dst[e];
  const float4 v = *(const float4*)(h + s * HIDDEN + q);
  float* o = outbuf + d * HIDDEN + q;
  atomicAdd(o + 0, v.x);
  atomicAdd(o + 1, v.y);
  atomicAdd(o + 2, v.z);
  atomicAdd(o + 3, v.w);
}

// K3: per 16-node tile: C = relu(out_tile) @ W2 via V_WMMA_F32_16X16X4_F32,
// then relu(C + b2) column-summed into acc (for the mean).
__global__ void k3_wmma_reduce(const float* __restrict__ outbuf,
                               const float* __restrict__ W2,
                               const float* __restrict__ b2,
                               float* __restrict__ acc,
                               int n_tiles) {
  const int lane   = threadIdx.x & 31;
  const int wid    = (blockIdx.x * blockDim.x + threadIdx.x) >> 5;
  const int nwaves = (gridDim.x * blockDim.x) >> 5;
  const int n      = lane & 15;               // output column within half
  const int khalf  = (lane >> 4) << 1;        // lanes 0-15: +0 ; lanes 16-31: +2

  // Preload B = W2 (32x32, K-major) for both N-halves; invariant across tiles.
  v2f B0[8], B1[8];
#pragma unroll
  for (int kc = 0; kc < 8; ++kc) {
    int kb = 4 * kc + khalf;
    B0[kc].x = W2[(kb + 0) * HIDDEN + n];
    B0[kc].y = W2[(kb + 1) * HIDDEN + n];
    B1[kc].x = W2[(kb + 0) * HIDDEN + 16 + n];
    B1[kc].y = W2[(kb + 1) * HIDDEN + 16 + n];
  }
  const float bb0 = b2[n];
  const float bb1 = b2[16 + n];
  float s0 = 0.0f, s1 = 0.0f;

  for (int t = wid; t < n_tiles; t += nwaves) {   // wave-uniform: EXEC stays all-1s
    const float* tile = outbuf + (long long)t * 16 * HIDDEN;
    v8f c0 = {}; v8f c1 = {};
#pragma unroll
    for (int kc = 0; kc < 8; ++kc) {
      int kb = 4 * kc + khalf;
      const float* ap = tile + (lane & 15) * HIDDEN + kb;
      v2f a;
      a.x = fmaxf(ap[0], 0.0f);                   // relu(out)
      a.y = fmaxf(ap[1], 0.0f);
      c0 = __builtin_amdgcn_wmma_f32_16x16x4_f32(false, a, false, B0[kc],
                                                 (short)0, c0, false, false);
      c1 = __builtin_amdgcn_wmma_f32_16x16x4_f32(false, a, false, B1[kc],
                                                 (short)0, c1, false, false);
    }
#pragma unroll
    for (int r = 0; r < 8; ++r) {
      s0 += fmaxf(c0[r] + bb0, 0.0f);
      s1 += fmaxf(c1[r] + bb1, 0.0f);
    }
  }
  atomicAdd(&acc[n],      s0);
  atomicAdd(&acc[16 + n], s1);
}

// K3b: scalar tail for nodes not covered by full 16-row tiles (normally none).
__global__ void k3_tail(const float* __restrict__ outbuf,
                        const float* __restrict__ W2,
                        const float* __restrict__ b2,
                        float* __restrict__ acc,
                        int start, int n_nodes) {
  int idx = blockIdx.x * blockDim.x + threadIdx.x;
  int cnt = (n_nodes - start) * HIDDEN;
  if (idx >= cnt) return;
  int i = start + idx / HIDDEN;
  int j = idx & 31;
  float s = b2[j];
  for (int k = 0; k < HIDDEN; ++k)
    s += fmaxf(outbuf[(long long)i * HIDDEN + k], 0.0f) * W2[k * HIDDEN + j];
  atomicAdd(&acc[j], fmaxf(s, 0.0f));
}

// K4: mean, target MLP, prediction head, sigmoid. One wave (32 threads).
__global__ void k4_head(const float* __restrict__ acc,
                        const float* __restrict__ tfeat,
                        const float* __restrict__ tW1, const float* __restrict__ tb1,
                        const float* __restrict__ tW2, const float* __restrict__ tb2,
                        const float* __restrict__ pW1, const float* __restrict__ pb1,
                        const float* __restrict__ pW2, const float* __restrict__ pb2,
                        float* __restrict__ out, float inv_n) {
  __shared__ float z[64];
  __shared__ float t1[32];
  __shared__ float ph[32];
  int j = threadIdx.x;            // 0..31
  z[j] = acc[j] * inv_n;          // drug_emb
  float s = tb1[j];
#pragma unroll
  for (int k = 0; k < 8; ++k) s += tfeat[k] * tW1[k * 32 + j];
  t1[j] = fmaxf(s, 0.0f);
  __syncthreads();
  float s2 = tb2[j];
#pragma unroll
  for (int k = 0; k < 32; ++k) s2 += t1[k] * tW2[k * 32 + j];
  z[32 + j] = s2;                 // target_emb
  __syncthreads();
  float s3 = pb1[j];
#pragma unroll
  for (int k = 0; k < 64; ++k) s3 += z[k] * pW1[k * 32 + j];
  ph[j] = fmaxf(s3, 0.0f);
  __syncthreads();
  if (j == 0) {
    float logit = pb2[0];
#pragma unroll
    for (int k = 0; k < 32; ++k) logit += ph[k] * pW2[k];
    out[0] = 1.0f / (1.0f + expf(-logit));
  }
}

extern "C" void kernel_launch(void* const* d_in, const int* in_sizes, int n_in,
                              void* d_out, int out_size, void* d_ws, size_t ws_size,
                              hipStream_t stream) {
  const float* x     = (const float*)d_in[0];
  const int*   ei    = (const int*)d_in[1];
  const float* tfeat = (const float*)d_in[2];
  const float* dW1   = (const float*)d_in[3];
  const float* db1   = (const float*)d_in[4];
  const float* dW2   = (const float*)d_in[5];
  const float* db2   = (const float*)d_in[6];
  const float* tW1   = (const float*)d_in[7];
  const float* tb1   = (const float*)d_in[8];
  const float* tW2   = (const float*)d_in[9];
  const float* tb2   = (const float*)d_in[10];
  const float* pW1   = (const float*)d_in[11];
  const float* pb1   = (const float*)d_in[12];
  const float* pW2   = (const float*)d_in[13];
  const float* pb2   = (const float*)d_in[14];

  const long long n_nodes = (long long)in_sizes[0] / DFEAT;
  const long long n_edges = (long long)in_sizes[1] / 2;

  float* ws  = (float*)d_ws;
  float* h   = ws;                                   // N*32 floats
  float* ob  = ws + n_nodes * HIDDEN;                // N*32 floats
  float* acc = ob + n_nodes * HIDDEN;                // 32 floats

  // K1: node linear + out init + acc zero
  long long t1 = n_nodes * HIDDEN;
  int g1 = (int)((t1 + 255) / 256);
  k1_node_linear<<<g1, 256, 0, stream>>>(x, dW1, db1, h, ob, acc, t1);

  // K2: edge scatter-add (8 lanes / edge)
  long long t2 = n_edges * 8;
  int g2 = (int)((t2 + 255) / 256);
  k2_edge_scatter<<<g2, 256, 0, stream>>>(ei, ei + n_edges, h, ob, n_edges);

  // K3: WMMA f32 GEMM + relu + column-sum into acc
  int n_tiles = (int)(n_nodes / 16);
  if (n_tiles > 0)
    k3_wmma_reduce<<<512, 256, 0, stream>>>(ob, dW2, db2, acc, n_tiles);
  int tail_start = n_tiles * 16;
  if (tail_start < (int)n_nodes) {
    int cnt = ((int)n_nodes - tail_start) * HIDDEN;
    k3_tail<<<(cnt + 255) / 256, 256, 0, stream>>>(ob, dW2, db2, acc,
                                                   tail_start, (int)n_nodes);
  }

  // K4: mean + target path + prediction head + sigmoid
  k4_head<<<1, 32, 0, stream>>>(acc, tfeat, tW1, tb1, tW2, tb2,
                                pW1, pb1, pW2, pb2,
                                (float*)d_out, 1.0f / (float)n_nodes);
}